// GNNLayer_9887014715394
// MI455X (gfx1250) — compile-verified
//
#include <hip/hip_runtime.h>

// GNN layer: out = (SpMM(x) + x) @ W1^T + SpMM(x*x) @ W2^T + b1 + b2
// N = 100000 nodes, E = 3.2M edges, F = 128 features.

#define F 128

typedef __attribute__((ext_vector_type(2))) float v2f;
typedef __attribute__((ext_vector_type(8))) float v8f;

// ---------------------------------------------------------------------------
// Kernel 1: fused edge pass. One edge per wave32; lane l owns features
// [4l, 4l+4). Loads x[col] once (b128), forms both val*x and val*x^2
// messages, scatter-adds with hardware f32 atomics (L2-resident accumulators).
// ---------------------------------------------------------------------------
__global__ __launch_bounds__(256) void spmm_edge_kernel(
    const float* __restrict__ x,
    const int* __restrict__ erow,
    const int* __restrict__ ecol,
    const float* __restrict__ eval,
    float* __restrict__ acc1,
    float* __restrict__ acc2,
    int nEdges) {
  const int lane = threadIdx.x & 31;
  const int wave = (int)((blockIdx.x * blockDim.x + threadIdx.x) >> 5);
  const int nWaves = (int)((gridDim.x * blockDim.x) >> 5);
  const int fOff = lane * 4;

  for (int e = wave; e < nEdges; e += nWaves) {
    const int r = erow[e];
    const int c = ecol[e];
    const float v = eval[e];

    const float4 xv =
        *reinterpret_cast<const float4*>(x + (size_t)c * F + fOff);

    float* p1 = acc1 + (size_t)r * F + fOff;
    float* p2 = acc2 + (size_t)r * F + fOff;

    // msg1 = v * x, msg2 = v * x*x  (both terms from one gather)
    unsafeAtomicAdd(p1 + 0, v * xv.x);
    unsafeAtomicAdd(p1 + 1, v * xv.y);
    unsafeAtomicAdd(p1 + 2, v * xv.z);
    unsafeAtomicAdd(p1 + 3, v * xv.w);
    unsafeAtomicAdd(p2 + 0, v * xv.x * xv.x);
    unsafeAtomicAdd(p2 + 1, v * xv.y * xv.y);
    unsafeAtomicAdd(p2 + 2, v * xv.z * xv.z);
    unsafeAtomicAdd(p2 + 3, v * xv.w * xv.w);
  }
}

// ---------------------------------------------------------------------------
// Kernel 2: fused dual GEMM with V_WMMA_F32_16X16X4_F32.
//   out[m][n] = sum_k (acc1[m][k]+x[m][k]) * W1[n][k]
//            + sum_k  acc2[m][k]          * W2[n][k]  + b1[n] + b2[n]
// Block = 256 threads = 8 waves. Block owns 16 rows x 128 cols; wave w owns
// the 16x16 tile at columns [16w, 16w+16).
//
// WMMA f32 16x16x4 operand layout (wave32):
//   A (16x4): lane<16 -> M=lane, VGPR{0,1}=K{k0,k0+1}; lane>=16 -> K{k0+2,k0+3}
//   B (4x16): lane<16 -> N=lane, K pair {k0,k0+1};    lane>=16 -> {k0+2,k0+3}
//   C (16x16): VGPR r -> M = r + 8*(lane/16), N = lane%16
//
// W^T is staged into LDS pair-interleaved: lds[(k/2)*128 + n]*2 + (k&1),
// so each lane's B fragment is one aligned ds_load_b64. K sliced in halves
// of 64 so LDS = 2 matrices * 64*128 floats = 64 KB exactly.
// ---------------------------------------------------------------------------
__global__ __launch_bounds__(256) void fused_wmma_gemm_kernel(
    const float* __restrict__ x,
    const float* __restrict__ acc1,
    const float* __restrict__ acc2,
    const float* __restrict__ W1,  // [F][F] row-major: W1[out][in]
    const float* __restrict__ b1,
    const float* __restrict__ W2,
    const float* __restrict__ b2,
    float* __restrict__ out) {
  __shared__ float ldsW[2 * 32 * F * 2];  // 16384 floats = 64 KB
  constexpr int MAT2 = 32 * F * 2;        // offset of W2 slice

  const int tid = threadIdx.x;
  const int lane = tid & 31;
  const int waveId = tid >> 5;   // 0..7 -> which 16-col tile
  const int mHalf = lane >> 4;   // 0/1: selects K pair within the 4-wide step
  const int mIdx = lane & 15;    // M for A-frag, N for B-frag/C
  const int rowBase = (int)blockIdx.x * 16;
  const int col = waveId * 16 + mIdx;  // this lane's output column

  v8f c = {0.f, 0.f, 0.f, 0.f, 0.f, 0.f, 0.f, 0.f};

  for (int kb = 0; kb < F; kb += 64) {
    __syncthreads();  // protect previous slice's reads
    // Stage K-slice [kb, kb+64) of W1^T and W2^T, pair-interleaved.
    for (int idx = tid; idx < 64 * F; idx += 256) {
      const int kk = idx & 63;  // local k (fastest -> coalesced global reads)
      const int o = idx >> 6;   // weight row = output feature = B's N
      const int d = (((kk >> 1) * F + o) << 1) + (kk & 1);
      ldsW[d] = W1[o * F + kb + kk];
      ldsW[MAT2 + d] = W2[o * F + kb + kk];
    }
    __syncthreads();

    for (int k0 = 0; k0 < 64; k0 += 4) {
      // A fragments: per-lane float2 at column k0 + 2*mHalf of row rowBase+mIdx
      const int kA = kb + k0 + 2 * mHalf;
      const size_t rowOff = (size_t)(rowBase + mIdx) * F + kA;
      v2f a1 = *reinterpret_cast<const v2f*>(acc1 + rowOff);
      const v2f ax = *reinterpret_cast<const v2f*>(x + rowOff);
      a1 += ax;  // (L@x + x) term
      const v2f a2 = *reinterpret_cast<const v2f*>(acc2 + rowOff);

      // B fragments from LDS (one b64 per matrix)
      const int pLoc = (k0 >> 1) + mHalf;  // K-pair index within slice
      const v2f bf1 =
          *reinterpret_cast<const v2f*>(ldsW + ((pLoc * F + col) << 1));
      const v2f bf2 =
          *reinterpret_cast<const v2f*>(ldsW + MAT2 + ((pLoc * F + col) << 1));

      c = __builtin_amdgcn_wmma_f32_16x16x4_f32(false, a1, false, bf1,
                                                (short)0, c, false, false);
      c = __builtin_amdgcn_wmma_f32_16x16x4_f32(false, a2, false, bf2,
                                                (short)0, c, false, false);
    }
  }

  const float bias = b1[col] + b2[col];
#pragma unroll
  for (int r = 0; r < 8; ++r) {
    out[(size_t)(rowBase + r + 8 * mHalf) * F + col] = c[r] + bias;
  }
}

// ---------------------------------------------------------------------------
extern "C" void kernel_launch(void* const* d_in, const int* in_sizes, int n_in,
                              void* d_out, int out_size, void* d_ws,
                              size_t ws_size, hipStream_t stream) {
  const float* features = (const float*)d_in[0];
  const int* erow = (const int*)d_in[1];
  const int* ecol = (const int*)d_in[2];
  const float* eval = (const float*)d_in[3];
  const float* W1 = (const float*)d_in[4];
  const float* b1 = (const float*)d_in[5];
  const float* W2 = (const float*)d_in[6];
  const float* b2 = (const float*)d_in[7];
  float* out = (float*)d_out;

  const int nNodes = in_sizes[0] / F;  // 100000
  const int nEdges = in_sizes[1];      // 3200000

  float* acc1 = (float*)d_ws;
  float* acc2 = acc1 + (size_t)nNodes * F;

  // Zero the SpMM accumulators (graph-capturable memset node).
  hipMemsetAsync(d_ws, 0, (size_t)2 * nNodes * F * sizeof(float), stream);

  // Edge pass: 2048 blocks * 8 waves = 16384 waves, ~195 edges each.
  spmm_edge_kernel<<<2048, 256, 0, stream>>>(features, erow, ecol, eval, acc1,
                                             acc2, nEdges);

  // Fused dual-GEMM: one block per 16 output rows (100000/16 = 6250).
  const int mTiles = nNodes / 16;
  fused_wmma_gemm_kernel<<<mTiles, 256, 0, stream>>>(features, acc1, acc2, W1,
                                                     b1, W2, b2, out);
}